// TensorProduct_63634235457616
// MI455X (gfx1250) — compile-verified
//
#include <hip/hip_runtime.h>

// CDNA5 / gfx1250 tensor-product kernel.
// Store-bound problem (85MB out vs 1.2MB in): compute each path's 16x16
// multiplicity block per batch element with V_WMMA_F32_16X16X4_F32,
// descriptors via wave-uniform (scalar) loads, operand gathers from LDS,
// stage D in LDS, store coalesced.

typedef __attribute__((ext_vector_type(2))) float v2f;
typedef __attribute__((ext_vector_type(8))) float v8f;

#define NCOMBO 19
#define DIM_IN 144
#define SLOT   100   // ints per (combo,kc) descriptor: 1 count + 32*(ic,jc,cv)
#define MAXPOS 32    // max CG nonzeros per (combo,kc) (true max is 25)

// (l1, l2, l_out) paths in generation order; bases are the sorted-output
// block offsets (stable sort by l_out), verified to sum to dim_out = 20736.
__device__ __constant__ int C_L1[NCOMBO]   = {0,0,0,1,1,1,1,1,1,1,2,2,2,2,2,2,2,2,2};
__device__ __constant__ int C_L2[NCOMBO]   = {0,1,2,0,1,1,1,2,2,2,0,1,1,1,2,2,2,2,2};
__device__ __constant__ int C_LO[NCOMBO]   = {0,1,2,1,0,1,2,1,2,3,2,1,2,3,0,1,2,3,4};
__device__ __constant__ int C_BASE[NCOMBO] = {0,768,5376,1536,256,2304,6656,3072,7936,
                                              13056,9216,3840,10496,14848,512,4608,
                                              11776,16640,18432};
__device__ __constant__ int MOFF[3] = {0, 16, 64};   // input offset of l-block

// ---------------------------------------------------------------------------
// Setup: single thread extracts, for every (path, kc), the list of
// (ic, jc, cv) CG nonzeros from the w=0 slice of the sparse index arrays,
// zero-padded to K=4 chunks. Deterministic; rebuilt every call.
// ---------------------------------------------------------------------------
__global__ void tp_setup_kernel(const int* __restrict__ i_idx,
                                const int* __restrict__ j_idx,
                                const int* __restrict__ k_idx,
                                const float* __restrict__ vals,
                                int nnz, int* __restrict__ wsI) {
    if (threadIdx.x != 0 || blockIdx.x != 0) return;
    float* wsF = (float*)wsI;
    const int total = NCOMBO * 9 * SLOT;
    for (int t = 0; t < total; ++t) wsI[t] = 0;

    long s = 0;
    for (int c = 0; c < NCOMBO; ++c) {
        const int l1 = C_L1[c], l2 = C_L2[c], lo = C_LO[c];
        const int d1 = 2 * l1 + 1, d2 = 2 * l2 + 1, dl = 2 * lo + 1;
        const int i1 = MOFF[l1], i2 = MOFF[l2], base = C_BASE[c];
        // g = CG nonzeros for this path = length of the (u=0,v=0) run
        int g = 0;
        while (s + g < nnz) {
            unsigned a = (unsigned)(i_idx[s + g] - i1);
            unsigned b = (unsigned)(j_idx[s + g] - i2);
            if (a < (unsigned)d1 && b < (unsigned)d2) ++g; else break;
        }
        int cnt[9];
        for (int k = 0; k < 9; ++k) cnt[k] = 0;
        for (int e = 0; e < g; ++e) {
            int kc = k_idx[s + e] - base;          // w=0 -> row = base + kc
            if (kc < 0 || kc >= dl) continue;       // defensive
            int ic = i_idx[s + e] - i1;
            int jc = j_idx[s + e] - i2;
            float cv = vals[s + e];
            int pos = cnt[kc]++;
            if (pos < MAXPOS) {
                int b0 = (c * 9 + kc) * SLOT + 1 + pos * 3;
                wsI[b0 + 0] = ic;
                wsI[b0 + 1] = jc;
                wsF[b0 + 2] = cv;
            }
        }
        for (int kc = 0; kc < dl; ++kc) {
            int n = cnt[kc] < MAXPOS ? cnt[kc] : MAXPOS;
            wsI[(c * 9 + kc) * SLOT] = (n + 3) >> 2;   // K=4 chunks
        }
        s += 256L * g;   // skip all 256 (u,v) repeats of this path
    }
}

// ---------------------------------------------------------------------------
// Main: one wave32 per (z, path). For each output component kc, accumulate
// D[u,v] = sum_p cv_p * x1[z,u,ic_p] * x2[z,v,jc_p] via f32 WMMA 16x16x4.
// A layout (ISA 7.12.2): lanes 0-15 hold M=lane, K=0,1; lanes 16-31 M=lane-16,
// K=2,3. B mirrored: VGPR0 = rows K0 (lanes 0-15) / K2 (lanes 16-31), VGPR1 =
// K1/K3. D: VGPR r = row M=r (lanes 0-15) / M=r+8 (lanes 16-31), N=lane%16.
// Descriptors are read with wave-uniform indices (-> scalar loads); per-lane
// K-slot selection done with lane<16 ternaries. Input slices live in LDS.
// ---------------------------------------------------------------------------
__global__ __launch_bounds__(32)
void tp_wmma_kernel(const float* __restrict__ x1, const float* __restrict__ x2,
                    const int* __restrict__ wsI, float* __restrict__ out,
                    int dim_out) {
    __shared__ float sx1[80];                    // 16 x d1 slice of x1[z]
    __shared__ float sx2[80];                    // 16 x d2 slice of x2[z]
    __shared__ __align__(16) float so[2304];     // max 256*(2*4+1) D staging
    const float* wsF = (const float*)wsI;

    const int z = blockIdx.x;
    const int c = blockIdx.y;
    const int l1 = C_L1[c], l2 = C_L2[c], lo = C_LO[c];
    const int d1 = 2 * l1 + 1, d2 = 2 * l2 + 1, dl = 2 * lo + 1;
    const int i1 = MOFF[l1], i2 = MOFF[l2], base = C_BASE[c];

    const int  lane = threadIdx.x;
    const int  m    = lane & 15;                 // A row (u) / B col (v)
    const bool hi   = (lane >= 16);              // lane's K slots: hi->2,3

    // Cooperative coalesced preload of the two input slices
    const int nx1 = 16 * d1, nx2 = 16 * d2;
    const float* gx1 = x1 + (long)z * DIM_IN + i1;
    const float* gx2 = x2 + (long)z * DIM_IN + i2;
    for (int t = lane; t < nx1; t += 32) sx1[t] = gx1[t];
    for (int t = lane; t < nx2; t += 32) sx2[t] = gx2[t];
    __syncthreads();

    const int r1 = m * d1;                       // this lane's x1 row base
    const int r2 = m * d2;                       // this lane's x2 row base

    for (int kc = 0; kc < dl; ++kc) {
        const int sbase = (c * 9 + kc) * SLOT;
        const int nch = wsI[sbase];
        v8f acc = {0.f, 0.f, 0.f, 0.f, 0.f, 0.f, 0.f, 0.f};
        for (int ch = 0; ch < nch; ++ch) {
            // All 12 words fetched with wave-uniform indices -> s_load
            const int* cp = wsI + sbase + 1 + ch * 12;
            const float* cf = wsF + sbase + 1 + ch * 12;
            const int   ia0 = cp[0], ja0 = cp[1];
            const float cv0 = cf[2];
            const int   ia1 = cp[3], ja1 = cp[4];
            const float cv1 = cf[5];
            const int   ia2 = cp[6], ja2 = cp[7];
            const float cv2 = cf[8];
            const int   ia3 = cp[9], ja3 = cp[10];
            const float cv3 = cf[11];

            v2f a, b;
            a.x = (hi ? cv2 : cv0) * sx1[r1 + (hi ? ia2 : ia0)];
            a.y = (hi ? cv3 : cv1) * sx1[r1 + (hi ? ia3 : ia1)];
            b.x = sx2[r2 + (hi ? ja2 : ja0)];
            b.y = sx2[r2 + (hi ? ja3 : ja1)];
            acc = __builtin_amdgcn_wmma_f32_16x16x4_f32(
                false, a, false, b, (short)0, acc, false, false);
        }
        // Scatter D into LDS at the combo-local layout (w*dl + kc)
#pragma unroll
        for (int r = 0; r < 8; ++r) {
            int u = r + (hi ? 8 : 0);
            so[(u * 16 + m) * dl + kc] = acc[r];
        }
    }
    __syncthreads();

    // Coalesced float4 stores of the whole 256*dl contiguous block
    float* OUT = out + (long)z * dim_out + base;
    const int total = 256 * dl;                  // multiple of 4
    for (int off = lane * 4; off < total; off += 128) {
        float4 t = *(const float4*)&so[off];
        *(float4*)&OUT[off] = t;
    }
}

// ---------------------------------------------------------------------------
extern "C" void kernel_launch(void* const* d_in, const int* in_sizes, int n_in,
                              void* d_out, int out_size, void* d_ws, size_t ws_size,
                              hipStream_t stream) {
    const float* x1   = (const float*)d_in[0];
    const float* x2   = (const float*)d_in[1];
    const float* vals = (const float*)d_in[2];
    const int*   kix  = (const int*)d_in[3];
    const int*   iix  = (const int*)d_in[4];
    const int*   jix  = (const int*)d_in[5];
    (void)n_in; (void)ws_size;

    const int nnz = in_sizes[2];
    const int n   = in_sizes[0] / DIM_IN;        // batch size
    const int dim_out = out_size / n;            // 20736 for this RS

    int* wsI = (int*)d_ws;
    tp_setup_kernel<<<1, 1, 0, stream>>>(iix, jix, kix, vals, nnz, wsI);

    dim3 grid(n, NCOMBO);
    tp_wmma_kernel<<<grid, 32, 0, stream>>>(x1, x2, wsI, (float*)d_out, dim_out);
}